// SelfAttention_36146444763964
// MI455X (gfx1250) — compile-verified
//
#include <hip/hip_runtime.h>
#include <hip/hip_bf16.h>

// N=2, C=256, D,H,W = 8,64,64 -> M = 32768 tokens per batch.
// Channel attention factorized through the 256x256 Gram matrix G = X X^T.

#define MTOK 32768
#define CCH  256

typedef __attribute__((ext_vector_type(2))) float v2f;
typedef __attribute__((ext_vector_type(8))) float v8f;

__device__ __forceinline__ v8f wmma_f32(v2f a, v2f b, v8f c) {
    // V_WMMA_F32_16X16X4_F32 : D = A(16x4,f32) * B(4x16,f32) + C(16x16,f32)
    return __builtin_amdgcn_wmma_f32_16x16x4_f32(false, a, false, b, (short)0, c, false, false);
}

// ---------------------------------------------------------------------------
// Kernel 1: per-workgroup partial Gram matrix G += X X^T over a 512-token
// chunk, plus partial row-sums s = X*1.  16 waves; each wave owns a 4x4 block
// of 16x16 tiles (full 256x256).  X slab staged in LDS, pad=34 so the lane
// 16..31 fragment reads hit disjoint banks from lanes 0..15.
// ---------------------------------------------------------------------------
__global__ __launch_bounds__(512) void gram_kernel(const float* __restrict__ x,
                                                   float* __restrict__ Gpart,
                                                   float* __restrict__ Spart) {
    __shared__ float lds[CCH * 34];
    const int n    = blockIdx.y;
    const int wg   = blockIdx.x;              // 0..63, 512 tokens each
    const float* xn = x + (size_t)n * CCH * MTOK;
    const int tid  = threadIdx.x;
    const int lane = tid & 31;
    const int w    = tid >> 5;                // wave 0..15
    const int bwr  = w >> 2, bwc = w & 3;     // 4x4 grid of wave blocks
    const int l15  = lane & 15;
    const int half = lane >> 4;

    const v8f zero8 = {0.f,0.f,0.f,0.f,0.f,0.f,0.f,0.f};
    v8f acc[4][4];
#pragma unroll
    for (int r = 0; r < 4; ++r)
#pragma unroll
        for (int c = 0; c < 4; ++c) acc[r][c] = zero8;

    float sacc[16];
#pragma unroll
    for (int i = 0; i < 16; ++i) sacc[i] = 0.f;

    const int m0 = wg * 512;
    for (int t0 = 0; t0 < 512; t0 += 32) {
        __syncthreads();
        // Cooperative coalesced load of 256ch x 32tok slab; fold row-sum.
#pragma unroll
        for (int i = 0; i < 16; ++i) {
            const int ch = w + 16 * i;
            const float val = xn[(size_t)ch * MTOK + m0 + t0 + lane];
            lds[ch * 34 + lane] = val;
            sacc[i] += val;
        }
        __syncthreads();
#pragma unroll
        for (int kb = 0; kb < 32; kb += 4) {
            const int kk = kb + half * 2;     // K pair for this lane half
            v2f a[4], b[4];
#pragma unroll
            for (int r = 0; r < 4; ++r) {
                const int chA = (bwr * 4 + r) * 16 + l15;
                a[r] = *(const v2f*)&lds[chA * 34 + kk];
                const int chB = (bwc * 4 + r) * 16 + l15;
                b[r] = *(const v2f*)&lds[chB * 34 + kk];
            }
#pragma unroll
            for (int r = 0; r < 4; ++r)
#pragma unroll
                for (int c = 0; c < 4; ++c)
                    acc[r][c] = wmma_f32(a[r], b[c], acc[r][c]);
        }
    }

    float* gp = Gpart + ((size_t)n * 64 + wg) * (CCH * CCH);
#pragma unroll
    for (int r = 0; r < 4; ++r)
#pragma unroll
        for (int c = 0; c < 4; ++c) {
            const int row0 = (bwr * 4 + r) * 16 + half * 8;
            const int col  = (bwc * 4 + c) * 16 + l15;
#pragma unroll
            for (int v = 0; v < 8; ++v)
                gp[(row0 + v) * CCH + col] = acc[r][c][v];
        }

    // deterministic wave reduction of row-sums
#pragma unroll
    for (int i = 0; i < 16; ++i) {
        float s = sacc[i];
        for (int off = 16; off >= 1; off >>= 1) s += __shfl_xor(s, off, 32);
        if (lane == 0) Spart[((size_t)n * 64 + wg) * CCH + w + 16 * i] = s;
    }
}

// ---------------------------------------------------------------------------
// Kernel 2: tree-reduce partials -> G[n], s[n]  (fixed order: deterministic)
// ---------------------------------------------------------------------------
__global__ __launch_bounds__(256) void reduce_kernel(const float* __restrict__ Gpart,
                                                     const float* __restrict__ Spart,
                                                     float* __restrict__ G,
                                                     float* __restrict__ S) {
    const int idx = blockIdx.x * 256 + threadIdx.x;  // 0..131071
    const int n = idx >> 16;
    const int e = idx & 65535;
    const float* gp = Gpart + (size_t)n * 64 * 65536 + e;
    float s = 0.f;
    for (int p = 0; p < 64; ++p) s += gp[(size_t)p * 65536];
    G[idx] = s;
    if (idx < 512) {
        const float* sp = Spart + (size_t)(idx >> 8) * 64 * CCH + (idx & 255);
        float t = 0.f;
        for (int p = 0; p < 64; ++p) t += sp[p * CCH];
        S[idx] = t;
    }
}

// ---------------------------------------------------------------------------
// Kernel 3: rank-1 logit vectors  u = Wq s ;  w = Wk s + M*bk
// ---------------------------------------------------------------------------
__global__ __launch_bounds__(512) void vecprep_kernel(const float* __restrict__ Wq,
                                                      const float* __restrict__ Wk,
                                                      const float* __restrict__ bk,
                                                      const float* __restrict__ S,
                                                      float* __restrict__ U,
                                                      float* __restrict__ Wvec) {
    const int tid = threadIdx.x;                  // 0..511
    const int n = tid >> 8, i = tid & 255;
    const float* s = S + n * CCH;
    float u = 0.f, wv = 0.f;
    for (int c = 0; c < CCH; ++c) {
        u  += Wq[i * CCH + c] * s[c];
        wv += Wk[i * CCH + c] * s[c];
    }
    U[tid] = u;
    Wvec[tid] = wv + (float)MTOK * bk[i];
}

// ---------------------------------------------------------------------------
// Kernel 4 (generic 256x256x256 fp32 WMMA GEMM):  D = A * op(B)
//   transB=0 : D[i,j] = sum_k A[i,k] B[k,j]
//   transB=1 : D[i,j] = sum_k A[i,k] B[j,k]
// grid(4, N): each WG does 64 rows x 256 cols; wave -> 1 row-tile x 4 col-tiles.
// ---------------------------------------------------------------------------
__global__ __launch_bounds__(512) void gemm256_kernel(const float* __restrict__ A, long long sA,
                                                      const float* __restrict__ B, long long sB,
                                                      int transB,
                                                      float* __restrict__ D, long long sD) {
    const int n = blockIdx.y;
    const float* An = A + (size_t)n * sA;
    const float* Bn = B + (size_t)n * sB;
    float* Dn = D + (size_t)n * sD;
    const int tid = threadIdx.x, lane = tid & 31, w = tid >> 5;
    const int l15 = lane & 15, half = lane >> 4;
    const int row0 = blockIdx.x * 64;
    const int tr  = w & 3;
    const int tc0 = (w >> 2) * 4;
    const int rowA = row0 + tr * 16 + l15;

    const v8f zero8 = {0.f,0.f,0.f,0.f,0.f,0.f,0.f,0.f};
    v8f acc[4];
#pragma unroll
    for (int c = 0; c < 4; ++c) acc[c] = zero8;

    for (int kb = 0; kb < CCH; kb += 4) {
        const int kk = kb + half * 2;
        const v2f a = *(const v2f*)&An[rowA * CCH + kk];
#pragma unroll
        for (int c = 0; c < 4; ++c) {
            const int col = (tc0 + c) * 16 + l15;
            v2f b;
            if (transB) {
                b = *(const v2f*)&Bn[col * CCH + kk];
            } else {
                b[0] = Bn[kk * CCH + col];
                b[1] = Bn[(kk + 1) * CCH + col];
            }
            acc[c] = wmma_f32(a, b, acc[c]);
        }
    }
#pragma unroll
    for (int c = 0; c < 4; ++c) {
        const int col = (tc0 + c) * 16 + l15;
        const int rr = row0 + tr * 16 + half * 8;
#pragma unroll
        for (int v = 0; v < 8; ++v)
            Dn[(rr + v) * CCH + col] = acc[c][v];
    }
}

// ---------------------------------------------------------------------------
// Kernel 5: row softmax of logits P = P0 + u[i]*bk[j] + bq[i]*w[j],
// one row per wave (max-subtracted, logits have std ~ sqrt(M) ~ 181),
// also cvec[row] = att_row . bv
// ---------------------------------------------------------------------------
__global__ __launch_bounds__(256) void softmax_kernel(const float* __restrict__ P0,
                                                      const float* __restrict__ U,
                                                      const float* __restrict__ Wvec,
                                                      const float* __restrict__ bq,
                                                      const float* __restrict__ bk,
                                                      const float* __restrict__ bv,
                                                      float* __restrict__ att,
                                                      float* __restrict__ cvec) {
    const int n = blockIdx.y;
    const int row = blockIdx.x * 8 + (threadIdx.x >> 5);
    const int lane = threadIdx.x & 31;
    const float* p = P0 + ((size_t)n * CCH + row) * CCH;
    const float ui = U[n * CCH + row];
    const float bqi = bq[row];
    float l[8];
    float mx = -3.0e38f;
#pragma unroll
    for (int t = 0; t < 8; ++t) {
        const int j = lane + 32 * t;
        const float val = p[j] + ui * bk[j] + bqi * Wvec[n * CCH + j];
        l[t] = val;
        mx = fmaxf(mx, val);
    }
    for (int off = 16; off >= 1; off >>= 1) mx = fmaxf(mx, __shfl_xor(mx, off, 32));
    float sum = 0.f;
#pragma unroll
    for (int t = 0; t < 8; ++t) { l[t] = __expf(l[t] - mx); sum += l[t]; }
    for (int off = 16; off >= 1; off >>= 1) sum += __shfl_xor(sum, off, 32);
    const float inv = 1.f / sum;
    float cd = 0.f;
    float* arow = att + ((size_t)n * CCH + row) * CCH;
#pragma unroll
    for (int t = 0; t < 8; ++t) {
        const int j = lane + 32 * t;
        const float a = l[t] * inv;
        arow[j] = a;
        cd += a * bv[j];
    }
    for (int off = 16; off >= 1; off >>= 1) cd += __shfl_xor(cd, off, 32);
    if (lane == 0) cvec[n * CCH + row] = cd;
}

// ---------------------------------------------------------------------------
// Kernel 6: out[n,m,j] = sum_e x[n,e,m] * Bt[n,j,e] + cvec[n,j]
// 32 tokens per WG, X slab in LDS (pad=40 -> conflict-free transposed reads),
// wave -> 1 token-tile x 2 col-tiles, K over 256 channels.
// ---------------------------------------------------------------------------
__global__ __launch_bounds__(512) void out_kernel(const float* __restrict__ x,
                                                  const float* __restrict__ Bt,
                                                  const float* __restrict__ cvec,
                                                  float* __restrict__ out) {
    __shared__ float lds[CCH * 40];
    const int n = blockIdx.y;
    const float* xn = x + (size_t)n * CCH * MTOK;
    float* on = out + (size_t)n * MTOK * CCH;
    const int m0 = blockIdx.x * 32;
    const int tid = threadIdx.x, lane = tid & 31, w = tid >> 5;
    const int l15 = lane & 15, half = lane >> 4;

#pragma unroll
    for (int i = 0; i < 16; ++i) {
        const int ch = w + 16 * i;
        lds[ch * 40 + lane] = xn[(size_t)ch * MTOK + m0 + lane];
    }
    __syncthreads();

    const int tr  = w & 1;           // token tile (2 x 16 tokens)
    const int tc0 = (w >> 1) * 2;    // 2 col tiles
    const float* btn = Bt + (size_t)n * CCH * CCH;
    const v8f zero8 = {0.f,0.f,0.f,0.f,0.f,0.f,0.f,0.f};
    v8f acc[2];
    acc[0] = zero8; acc[1] = zero8;

    for (int kb = 0; kb < CCH; kb += 4) {
        const int kk = kb + half * 2;
        const int ml = tr * 16 + l15;
        v2f a;
        a[0] = lds[kk * 40 + ml];          // X^T fragment: A[m, e] = x[e, m]
        a[1] = lds[(kk + 1) * 40 + ml];
#pragma unroll
        for (int c = 0; c < 2; ++c) {
            const int col = (tc0 + c) * 16 + l15;
            const v2f b = *(const v2f*)&btn[col * CCH + kk];  // B[e,j] = Bt[j,e]
            acc[c] = wmma_f32(a, b, acc[c]);
        }
    }
#pragma unroll
    for (int c = 0; c < 2; ++c) {
        const int col = (tc0 + c) * 16 + l15;
        const float cv = cvec[n * CCH + col];
        const int mrow = tr * 16 + half * 8;
#pragma unroll
        for (int v = 0; v < 8; ++v)
            on[(size_t)(m0 + mrow + v) * CCH + col] = acc[c][v] + cv;
    }
}

// ---------------------------------------------------------------------------
extern "C" void kernel_launch(void* const* d_in, const int* in_sizes, int n_in,
                              void* d_out, int out_size, void* d_ws, size_t ws_size,
                              hipStream_t stream) {
    (void)in_sizes; (void)n_in; (void)out_size; (void)ws_size;
    const float* x  = (const float*)d_in[0];
    const float* Wq = (const float*)d_in[1];
    const float* bq = (const float*)d_in[2];
    const float* Wk = (const float*)d_in[3];
    const float* bk = (const float*)d_in[4];
    const float* Wv = (const float*)d_in[5];
    const float* bv = (const float*)d_in[6];
    float* out = (float*)d_out;
    float* ws  = (float*)d_ws;

    // workspace layout (floats)
    const size_t oGP  = 0;                                   // [2][64][256][256]
    const size_t oSP  = oGP  + (size_t)2 * 64 * 65536;       // [2][64][256]
    const size_t oG   = oSP  + (size_t)2 * 64 * 256;         // [2][256][256]
    const size_t oS   = oG   + (size_t)2 * 65536;            // [2][256]
    const size_t oT1  = oS   + 512;                          // [2][256][256]
    const size_t oP0  = oT1  + (size_t)2 * 65536;            // [2][256][256]
    const size_t oATT = oP0  + (size_t)2 * 65536;            // [2][256][256]
    const size_t oBT  = oATT + (size_t)2 * 65536;            // [2][256][256]
    const size_t oU   = oBT  + (size_t)2 * 65536;            // [2][256]
    const size_t oW   = oU   + 512;                          // [2][256]
    const size_t oCV  = oW   + 512;                          // [2][256]

    gram_kernel<<<dim3(64, 2), 512, 0, stream>>>(x, ws + oGP, ws + oSP);
    reduce_kernel<<<512, 256, 0, stream>>>(ws + oGP, ws + oSP, ws + oG, ws + oS);
    vecprep_kernel<<<1, 512, 0, stream>>>(Wq, Wk, bk, ws + oS, ws + oU, ws + oW);
    // T1 = Wq * G
    gemm256_kernel<<<dim3(4, 2), 512, 0, stream>>>(Wq, 0, ws + oG, 65536, 0, ws + oT1, 65536);
    // P0 = T1 * Wk^T
    gemm256_kernel<<<dim3(4, 2), 512, 0, stream>>>(ws + oT1, 65536, Wk, 0, 1, ws + oP0, 65536);
    softmax_kernel<<<dim3(32, 2), 256, 0, stream>>>(ws + oP0, ws + oU, ws + oW, bq, bk, bv,
                                                    ws + oATT, ws + oCV);
    // Bt = att * Wv
    gemm256_kernel<<<dim3(4, 2), 512, 0, stream>>>(ws + oATT, 65536, Wv, 0, 0, ws + oBT, 65536);
    out_kernel<<<dim3(1024, 2), 512, 0, stream>>>(x, ws + oBT, ws + oCV, out);
}